// KnowledgeIntegrationLoss_2980707303764
// MI455X (gfx1250) — compile-verified
//
#include <hip/hip_runtime.h>
#include <math.h>

typedef float v2f __attribute__((ext_vector_type(2)));
typedef float v8f __attribute__((ext_vector_type(8)));

#define KDIM   8192
#define N1     512
#define N2     256
#define N3     128
#define KSPLIT 32
#define KCHUNK (KDIM / KSPLIT)   // 256
#define EPS    1e-8f

// ---------------------------------------------------------------------------
// Kernel 1: layer-1 partial GEMM.  X[16,8192] (11 real rows, 5 zero) @ W1[8192,512].
// grid = (8 n-groups of 64 cols) x (32 k-chunks of 256). One wave32 per block.
// Each wave holds 4 WMMA accumulators (16x64 tile) and streams its K-chunk of W1
// exactly once. Partials -> Hp[KSPLIT][16][512].
// ---------------------------------------------------------------------------
__global__ void kil_layer1_partial(const float* __restrict__ xnew,
                                   const float* __restrict__ xex,
                                   const float* __restrict__ W1,
                                   float* __restrict__ Hp) {
    const int lane = threadIdx.x;       // 0..31
    const int half = lane >> 4;         // 0: K lo pair, 1: K hi pair
    const int l16  = lane & 15;
    const int ng   = blockIdx.x;        // n-group (64 cols)
    const int kc   = blockIdx.y;        // k-chunk

    // Virtual X row for A fragment: row 0 = new, 1..10 = existing, 11..15 = 0.
    const int m = l16;
    const float rowScale = (m <= 10) ? 1.0f : 0.0f;
    const float* xrow = (m == 0) ? xnew
                      : (m <= 10 ? (xex + (size_t)(m - 1) * KDIM) : xnew);

    v8f c0 = {0.f,0.f,0.f,0.f,0.f,0.f,0.f,0.f};
    v8f c1 = c0, c2 = c0, c3 = c0;

    const int k0 = kc * KCHUNK;
    const int nb = ng * 64 + l16;

    #pragma unroll 4
    for (int kk = 0; kk < KCHUNK; kk += 4) {
        const int kb = k0 + kk + half * 2;    // this lane-half's K base
        v2f a;
        a.x = xrow[kb]     * rowScale;        // A vgpr0: K = 2*half + 0
        a.y = xrow[kb + 1] * rowScale;        // A vgpr1: K = 2*half + 1

        v2f b0, b1, b2, b3;
        b0.x = W1[(kb    ) * N1 + nb +  0];  b0.y = W1[(kb + 1) * N1 + nb +  0];
        b1.x = W1[(kb    ) * N1 + nb + 16];  b1.y = W1[(kb + 1) * N1 + nb + 16];
        b2.x = W1[(kb    ) * N1 + nb + 32];  b2.y = W1[(kb + 1) * N1 + nb + 32];
        b3.x = W1[(kb    ) * N1 + nb + 48];  b3.y = W1[(kb + 1) * N1 + nb + 48];

        c0 = __builtin_amdgcn_wmma_f32_16x16x4_f32(false, a, false, b0, (short)0, c0, false, false);
        c1 = __builtin_amdgcn_wmma_f32_16x16x4_f32(false, a, false, b1, (short)0, c1, false, false);
        c2 = __builtin_amdgcn_wmma_f32_16x16x4_f32(false, a, false, b2, (short)0, c2, false, false);
        c3 = __builtin_amdgcn_wmma_f32_16x16x4_f32(false, a, false, b3, (short)0, c3, false, false);
    }

    // Store C tiles: VGPR v -> row (v + 8*half), col = lane%16 within subtile.
    float* outp = Hp + (size_t)kc * 16 * N1;
    #pragma unroll
    for (int v = 0; v < 8; ++v) {
        const int row = v + half * 8;
        outp[row * N1 + nb +  0] = c0[v];
        outp[row * N1 + nb + 16] = c1[v];
        outp[row * N1 + nb + 32] = c2[v];
        outp[row * N1 + nb + 48] = c3[v];
    }
}

// ---------------------------------------------------------------------------
// Kernel 2: reduce K-split partials + bias + ReLU -> H1[16,512]
// ---------------------------------------------------------------------------
__global__ void kil_reduce1(const float* __restrict__ Hp,
                            const float* __restrict__ b1,
                            float* __restrict__ H1) {
    const int idx = blockIdx.x * blockDim.x + threadIdx.x;   // 0..8191
    if (idx < 16 * N1) {
        const int n = idx & (N1 - 1);
        float s = 0.f;
        #pragma unroll 8
        for (int k = 0; k < KSPLIT; ++k) s += Hp[(size_t)k * 16 * N1 + idx];
        s += b1[n];
        H1[idx] = fmaxf(s, 0.f);
    }
}

// ---------------------------------------------------------------------------
// Generic small GEMM: out[16,N] = act(A[16,K] @ B[K,N] + bias). One wave32 per
// 16-col tile, f32 WMMA over K in steps of 4.
// ---------------------------------------------------------------------------
__global__ void kil_gemm16(const float* __restrict__ A,
                           const float* __restrict__ B,
                           const float* __restrict__ bias,
                           float* __restrict__ out,
                           int K, int N, int do_relu) {
    const int lane = threadIdx.x;
    const int half = lane >> 4;
    const int l16  = lane & 15;
    const int n    = blockIdx.x * 16 + l16;

    v8f c = {0.f,0.f,0.f,0.f,0.f,0.f,0.f,0.f};

    for (int kk = 0; kk < K; kk += 4) {
        const int kb = kk + half * 2;
        v2f a;
        a.x = A[l16 * K + kb];
        a.y = A[l16 * K + kb + 1];
        v2f b;
        b.x = B[(kb    ) * N + n];
        b.y = B[(kb + 1) * N + n];
        c = __builtin_amdgcn_wmma_f32_16x16x4_f32(false, a, false, b, (short)0, c, false, false);
    }

    const float bv = bias[n];
    #pragma unroll
    for (int v = 0; v < 8; ++v) {
        const int row = v + half * 8;
        float r = c[v] + bv;
        if (do_relu) r = fmaxf(r, 0.f);
        out[row * N + n] = r;
    }
}

// ---------------------------------------------------------------------------
// Final kernel: 1 block x 128 threads. Consistency terms + integrator MLP.
// E[16,128]: row 0 = new_enc, rows 1..10 = ex_enc.
// ---------------------------------------------------------------------------
__device__ __forceinline__ float kil_blk_reduce128(float v, float* red, int t) {
    red[t] = v; __syncthreads();
    #pragma unroll
    for (int s = 64; s > 0; s >>= 1) {
        if (t < s) red[t] += red[t + s];
        __syncthreads();
    }
    const float r = red[0];
    __syncthreads();
    return r;
}

__global__ void kil_final(const float* __restrict__ E,
                          const float* __restrict__ metrics,
                          const float* __restrict__ Wi1, const float* __restrict__ bi1,
                          const float* __restrict__ Wi2, const float* __restrict__ bi2,
                          const float* __restrict__ Wi3, const float* __restrict__ bi3,
                          float* __restrict__ out) {
    __shared__ float red[128];
    __shared__ float comb[134];
    __shared__ float h1[64];
    __shared__ float h2[32];

    const int t = threadIdx.x;           // 0..127
    const float nv = E[t];               // new_enc[t]

    const float ssn  = kil_blk_reduce128(nv * nv, red, t);
    const float newn = fmaxf(sqrtf(ssn), EPS);

    float acc = 0.f;
    for (int i = 0; i < 10; ++i) {
        const float ev = E[(1 + i) * N3 + t];
        const float d  = nv - ev;
        const float ssd = kil_blk_reduce128(d * d,   red, t);
        const float dot = kil_blk_reduce128(ev * nv, red, t);
        const float sse = kil_blk_reduce128(ev * ev, red, t);
        const float geo = sqrtf(ssd);
        const float exn = fmaxf(sqrtf(sse), EPS);
        acc += geo - dot / (newn * exn);
    }
    const float avg = acc * 0.1f;

    comb[t] = nv;
    if (t < 6) comb[128 + t] = metrics[t];
    __syncthreads();

    if (t < 64) {
        float s = bi1[t];
        for (int i = 0; i < 134; ++i) s += comb[i] * Wi1[i * 64 + t];
        h1[t] = fmaxf(s, 0.f);
    }
    __syncthreads();

    if (t < 32) {
        float s = bi2[t];
        for (int i = 0; i < 64; ++i) s += h1[i] * Wi2[i * 32 + t];
        h2[t] = fmaxf(s, 0.f);
    }
    __syncthreads();

    if (t == 0) {
        float q = bi3[0];
        for (int i = 0; i < 32; ++i) q += h2[i] * Wi3[i];
        out[0] = avg + expf(-q);
    }
}

// ---------------------------------------------------------------------------
extern "C" void kernel_launch(void* const* d_in, const int* in_sizes, int n_in,
                              void* d_out, int out_size, void* d_ws, size_t ws_size,
                              hipStream_t stream) {
    const float* xnew    = (const float*)d_in[0];   // [8192]
    const float* xex     = (const float*)d_in[1];   // [10,8192]
    const float* metrics = (const float*)d_in[2];   // [6]
    const float* W1 = (const float*)d_in[3];        // [8192,512]
    const float* b1 = (const float*)d_in[4];        // [512]
    const float* W2 = (const float*)d_in[5];        // [512,256]
    const float* b2 = (const float*)d_in[6];        // [256]
    const float* W3 = (const float*)d_in[7];        // [256,128]
    const float* b3 = (const float*)d_in[8];        // [128]
    const float* Wi1 = (const float*)d_in[9];       // [134,64]
    const float* bi1 = (const float*)d_in[10];      // [64]
    const float* Wi2 = (const float*)d_in[11];      // [64,32]
    const float* bi2 = (const float*)d_in[12];      // [32]
    const float* Wi3 = (const float*)d_in[13];      // [32,1]
    const float* bi3 = (const float*)d_in[14];      // [1]

    float* ws = (float*)d_ws;
    float* Hp = ws;                                 // KSPLIT*16*512 = 262144 f
    float* H1 = Hp + (size_t)KSPLIT * 16 * N1;      // 16*512
    float* H2 = H1 + 16 * N1;                       // 16*256
    float* Ee = H2 + 16 * N2;                       // 16*128

    dim3 g1(N1 / 64, KSPLIT);
    kil_layer1_partial<<<g1, 32, 0, stream>>>(xnew, xex, W1, Hp);
    kil_reduce1<<<(16 * N1 + 255) / 256, 256, 0, stream>>>(Hp, b1, H1);
    kil_gemm16<<<N2 / 16, 32, 0, stream>>>(H1, W2, b2, H2, N1, N2, 1);
    kil_gemm16<<<N3 / 16, 32, 0, stream>>>(H2, W3, b3, Ee, N2, N3, 0);
    kil_final<<<1, 128, 0, stream>>>(Ee, metrics, Wi1, bi1, Wi2, bi2, Wi3, bi3,
                                     (float*)d_out);
}